// FraudGCN_1709396983810
// MI455X (gfx1250) — compile-verified
//
#include <hip/hip_runtime.h>

#define CH 128

typedef float v2f __attribute__((ext_vector_type(2)));
typedef float v8f __attribute__((ext_vector_type(8)));

// ---------------------------------------------------------------------------
// Fill a float buffer with a constant (used for zero-init of scatter targets
// and for initializing degree to 1.0 to account for self-loops).
// ---------------------------------------------------------------------------
__global__ void fill_kernel(float* __restrict__ p, int n, float v) {
    int i = blockIdx.x * blockDim.x + threadIdx.x;
    if (i < n) p[i] = v;
}

// deg[dst[e]] += 1  (deg pre-filled with 1.0 for the self loop)
__global__ void deg_count_kernel(const int* __restrict__ dst,
                                 float* __restrict__ deg, int E) {
    int e = blockIdx.x * blockDim.x + threadIdx.x;
    if (e < E) atomicAdd(&deg[dst[e]], 1.0f);
}

__global__ void rsqrt_kernel(const float* __restrict__ deg,
                             float* __restrict__ dinv, int n) {
    int i = blockIdx.x * blockDim.x + threadIdx.x;
    if (i < n) dinv[i] = rsqrtf(deg[i]);
}

// ---------------------------------------------------------------------------
// FP32 GEMM via V_WMMA_F32_16X16X4_F32.
// H[M,128] = X[M,128] @ W[128,128], row-major, M divisible by 16.
// Block = 256 threads = 8 waves. Block b covers rows [16b,16b+16); wave w
// covers output cols [16w,16w+16). Per ISA layouts:
//   A (16x4 f32): lanes 0-15 -> M=lane, VGPR0=K0/VGPR1=K1; lanes 16-31 ->
//                 same M, K2/K3.  -> per-lane b64 load, contiguous.
//   B (4x16 f32): VGPR0 = row K0 (lanes 0-15, N=lane) / row K2 (lanes 16-31);
//                 VGPR1 = rows K1/K3.
//   C/D (16x16 f32): VGPR j: lanes 0-15 -> M=j, lanes 16-31 -> M=j+8, N=lane&15.
// ---------------------------------------------------------------------------
__global__ __launch_bounds__(256)
void gemm_wmma_f32(const float* __restrict__ X, const float* __restrict__ W,
                   float* __restrict__ H) {
    const int wave  = threadIdx.x >> 5;   // 0..7 -> N tile
    const int lane  = threadIdx.x & 31;
    const int m0    = blockIdx.x * 16;
    const int n0    = wave * 16;
    const int mrow  = lane & 15;          // row within A tile
    const int khalf = (lane >> 4) << 1;   // 0 or 2
    const int ncol  = lane & 15;          // col within B/C tile

    const float* __restrict__ xrow = X + (size_t)(m0 + mrow) * CH;

    v8f acc = {0.f, 0.f, 0.f, 0.f, 0.f, 0.f, 0.f, 0.f};
#pragma unroll
    for (int k0 = 0; k0 < CH; k0 += 4) {
        v2f a, b;
        a.x = xrow[k0 + khalf];
        a.y = xrow[k0 + khalf + 1];
        b.x = W[(size_t)(k0 + khalf) * CH + n0 + ncol];
        b.y = W[(size_t)(k0 + khalf + 1) * CH + n0 + ncol];
        acc = __builtin_amdgcn_wmma_f32_16x16x4_f32(
            /*neg_a=*/false, a, /*neg_b=*/false, b,
            /*c_mod=*/(short)0, acc, /*reuse_a=*/false, /*reuse_b=*/false);
    }

    const int moff = (lane >> 4) * 8;
#pragma unroll
    for (int j = 0; j < 8; ++j) {
        H[(size_t)(m0 + j + moff) * CH + n0 + ncol] = acc[j];
    }
}

// ---------------------------------------------------------------------------
// Edge aggregation, 128 channels: one wave per edge (incl. implicit self
// loops e >= E with src=dst=e-E). Lane loads float4 (b128) of h[src], scales
// by norm = dinv[src]*dinv[dst], atomically adds into out[dst]. Feature
// matrices are 51.2 MB -> L2 resident; atomics resolve at L2.
// ---------------------------------------------------------------------------
__global__ __launch_bounds__(256)
void scatter128_kernel(const float* __restrict__ h, const int* __restrict__ src,
                       const int* __restrict__ dst, const float* __restrict__ dinv,
                       float* __restrict__ out, int E, int Etot) {
    const int w = (blockIdx.x * blockDim.x + threadIdx.x) >> 5;  // wave32
    if (w >= Etot) return;
    const int lane = threadIdx.x & 31;
    int s, d;
    if (w < E) { s = src[w]; d = dst[w]; } else { s = d = w - E; }
    const float nrm = dinv[s] * dinv[d];
    const float4 v = ((const float4*)(h + (size_t)s * CH))[lane];
    float* od = out + (size_t)d * CH + lane * 4;
    atomicAdd(od + 0, v.x * nrm);
    atomicAdd(od + 1, v.y * nrm);
    atomicAdd(od + 2, v.z * nrm);
    atomicAdd(od + 3, v.w * nrm);
}

// Fused  y = relu((aggr + bias - running_mean) * gamma/sqrt(rv+eps) + beta)
__global__ void postact_kernel(float* __restrict__ h, const float* __restrict__ b,
                               const float* __restrict__ g, const float* __restrict__ be,
                               const float* __restrict__ rm, const float* __restrict__ rv,
                               int total) {
    int i = blockIdx.x * blockDim.x + threadIdx.x;
    if (i >= total) return;
    const int c = i & (CH - 1);
    const float inv = g[c] * rsqrtf(rv[c] + 1e-5f);
    const float v = (h[i] + b[c] - rm[c]) * inv + be[c];
    h[i] = v > 0.f ? v : 0.f;
}

// Layer 3 linear transform: H3[M,2] = X[M,128] @ W3[128,2]  (51 MFLOP, VALU)
__global__ void gemm_out2_kernel(const float* __restrict__ X,
                                 const float* __restrict__ W3,
                                 float* __restrict__ H3, int M) {
    int i = blockIdx.x * blockDim.x + threadIdx.x;
    if (i >= M) return;
    const float* xr = X + (size_t)i * CH;
    float a0 = 0.f, a1 = 0.f;
#pragma unroll 8
    for (int k = 0; k < CH; ++k) {
        const float x = xr[k];
        a0 += x * W3[2 * k];
        a1 += x * W3[2 * k + 1];
    }
    H3[(size_t)i * 2]     = a0;
    H3[(size_t)i * 2 + 1] = a1;
}

// Edge aggregation, 2 channels: one thread per edge.
__global__ void scatter2_kernel(const float* __restrict__ h3, const int* __restrict__ src,
                                const int* __restrict__ dst, const float* __restrict__ dinv,
                                float* __restrict__ out, int E, int Etot) {
    int e = blockIdx.x * blockDim.x + threadIdx.x;
    if (e >= Etot) return;
    int s, d;
    if (e < E) { s = src[e]; d = dst[e]; } else { s = d = e - E; }
    const float nrm = dinv[s] * dinv[d];
    atomicAdd(&out[(size_t)d * 2],     h3[(size_t)s * 2]     * nrm);
    atomicAdd(&out[(size_t)d * 2 + 1], h3[(size_t)s * 2 + 1] * nrm);
}

__global__ void add_bias2_kernel(float* __restrict__ out,
                                 const float* __restrict__ b3, int total) {
    int i = blockIdx.x * blockDim.x + threadIdx.x;
    if (i < total) out[i] += b3[i & 1];
}

// ---------------------------------------------------------------------------
extern "C" void kernel_launch(void* const* d_in, const int* in_sizes, int n_in,
                              void* d_out, int out_size, void* d_ws, size_t ws_size,
                              hipStream_t stream) {
    const float* x   = (const float*)d_in[0];
    const int*   ei  = (const int*)d_in[1];
    const float* W1  = (const float*)d_in[2];
    const float* b1  = (const float*)d_in[3];
    const float* g1  = (const float*)d_in[4];
    const float* be1 = (const float*)d_in[5];
    const float* rm1 = (const float*)d_in[6];
    const float* rv1 = (const float*)d_in[7];
    const float* W2  = (const float*)d_in[8];
    const float* b2  = (const float*)d_in[9];
    const float* g2  = (const float*)d_in[10];
    const float* be2 = (const float*)d_in[11];
    const float* rm2 = (const float*)d_in[12];
    const float* rv2 = (const float*)d_in[13];
    const float* W3  = (const float*)d_in[14];
    const float* b3  = (const float*)d_in[15];

    const int Nn   = in_sizes[0] / CH;   // 100000
    const int Ee   = in_sizes[1] / 2;    // 1600000
    const int Etot = Ee + Nn;            // edges + self loops
    const int NC   = Nn * CH;            // 12.8M elements

    const int* src = ei;
    const int* dst = ei + Ee;

    // Workspace layout (floats): deg[N] dinv[N] bufA[N*128] bufB[N*128] h3[N*2]
    float* deg  = (float*)d_ws;
    float* dinv = deg  + Nn;
    float* bufA = dinv + Nn;
    float* bufB = bufA + (size_t)NC;
    float* h3   = bufB + (size_t)NC;
    float* out  = (float*)d_out;

    const int T = 256;
    const int gN  = (Nn + T - 1) / T;
    const int gE  = (Ee + T - 1) / T;
    const int gEt = (Etot + T - 1) / T;
    const int gNC = (NC + T - 1) / T;
    const int gSc = (Etot + 7) / 8;      // wave-per-edge, 8 waves/block
    const int gMM = Nn / 16;             // 6250 row tiles (exact)

    // Degree / normalization
    fill_kernel<<<gN, T, 0, stream>>>(deg, Nn, 1.0f);
    deg_count_kernel<<<gE, T, 0, stream>>>(dst, deg, Ee);
    rsqrt_kernel<<<gN, T, 0, stream>>>(deg, dinv, Nn);

    // Layer 1
    gemm_wmma_f32<<<gMM, T, 0, stream>>>(x, W1, bufA);
    fill_kernel<<<gNC, T, 0, stream>>>(bufB, NC, 0.0f);
    scatter128_kernel<<<gSc, T, 0, stream>>>(bufA, src, dst, dinv, bufB, Ee, Etot);
    postact_kernel<<<gNC, T, 0, stream>>>(bufB, b1, g1, be1, rm1, rv1, NC);

    // Layer 2
    gemm_wmma_f32<<<gMM, T, 0, stream>>>(bufB, W2, bufA);
    fill_kernel<<<gNC, T, 0, stream>>>(bufB, NC, 0.0f);
    scatter128_kernel<<<gSc, T, 0, stream>>>(bufA, src, dst, dinv, bufB, Ee, Etot);
    postact_kernel<<<gNC, T, 0, stream>>>(bufB, b2, g2, be2, rm2, rv2, NC);

    // Layer 3: transform first (2 channels), then cheap 2-wide scatter
    gemm_out2_kernel<<<gN, T, 0, stream>>>(bufB, W3, h3, Nn);
    fill_kernel<<<(2 * Nn + T - 1) / T, T, 0, stream>>>(out, 2 * Nn, 0.0f);
    scatter2_kernel<<<gEt, T, 0, stream>>>(h3, src, dst, dinv, out, Ee, Etot);
    add_bias2_kernel<<<(2 * Nn + T - 1) / T, T, 0, stream>>>(out, b3, 2 * Nn);
}